// GIN_30520037606037
// MI455X (gfx1250) — compile-verified
//
#include <hip/hip_runtime.h>

#define N_NODES   100000
#define N_EDGES   1600000
#define F_IN      128
#define HIDDEN    64
#define N_CLASSES 40
#define BN_EPS    1e-5f

typedef __attribute__((ext_vector_type(2))) float v2f;
typedef __attribute__((ext_vector_type(8))) float v8f;

// ---------------------------------------------------------------- zero ------
__global__ void gin_zero_kernel(float* __restrict__ p, int n4) {
    int i = blockIdx.x * blockDim.x + threadIdx.x;
    if (i < n4) {
        float4 z; z.x = 0.f; z.y = 0.f; z.z = 0.f; z.w = 0.f;
        ((float4*)p)[i] = z;
    }
}

// ----------------------------------------------------------- scatter-add ----
// agg[dst[e]][:] += h[src[e]][:]   (one thread per edge x 4-feature group)
template<int D>
__global__ void gin_scatter_kernel(const float* __restrict__ h,
                                   const long long* __restrict__ ei, // [2, E] int64
                                   float* __restrict__ agg) {
    constexpr int G = D / 4;                 // float4 groups per edge
    long long tid = (long long)blockIdx.x * blockDim.x + threadIdx.x;
    if (tid >= (long long)N_EDGES * G) return;
    int e = (int)(tid / G);
    int f = (int)(tid % G) * 4;
    __builtin_prefetch(ei + e + 2048, 0, 0); // global_prefetch_b8
    long long s = ei[e];
    long long t = ei[(long long)N_EDGES + e];
    const float4 v = *(const float4*)(h + (long long)s * D + f);
    float* a = agg + (long long)t * D + f;
    atomicAdd(a + 0, v.x);
    atomicAdd(a + 1, v.y);
    atomicAdd(a + 2, v.z);
    atomicAdd(a + 3, v.w);
}

// ------------------------------------------------- fused GEMM + BN + ReLU ---
// out[M,Nout] = act( ((h+agg)[M,K] @ W[K,Nout] + bias) BN? )
//
// Block = 256 threads (8 waves). W is staged TRANSPOSED into LDS once per
// block (Wt[n*K + k]) so each B fragment is one ds_load_b64 with an
// immediate offset. Each wave computes a 16-row tile; its A slice of
// (h+agg) is loaded as float2 pairs (global_load_b64, immediate offsets)
// and kept in registers across the N-tile loop.
//
// V_WMMA_F32_16X16X4_F32 operand layout (ISA 7.12.2):
//   A (16x4): lane l<16 holds rows M=l, {v0,v1}={K0,K1}; lanes 16-31 {K2,K3}
//   B (4x16): VGPR v holds row K=v+2*(lane>=16), col N=lane&15
//   C/D     : VGPR v holds row M=v+8*(lane>=16),  col N=lane&15
template<int K, bool USE_BN>
__launch_bounds__(256)
__global__ void gin_gemm_kernel(const float* __restrict__ h,
                                const float* __restrict__ agg,
                                const float* __restrict__ W,     // [K, Nout]
                                const float* __restrict__ bias,  // [Nout]
                                const float* __restrict__ gamma,
                                const float* __restrict__ beta,
                                const float* __restrict__ mean,
                                const float* __restrict__ var,
                                float* __restrict__ out,
                                int M, int Nout) {
    extern __shared__ float ldsWt[];         // [Nout, K] = W transposed

    // ---- cooperative transpose-fill of W into LDS (once per block)
    const int tid = threadIdx.x;
    for (int i = tid; i < K * Nout; i += 256) {
        const int k = i / Nout;              // coalesced read of W[k][n]
        const int n = i - k * Nout;
        ldsWt[n * K + k] = W[i];
    }
    __syncthreads();

    const int wave  = tid >> 5;
    const int lane  = tid & 31;
    const int mtile = blockIdx.x * 8 + wave;
    const int m0    = mtile * 16;
    if (m0 + 16 > M) return;                 // wave-uniform (after barrier)

    const int hi   = lane >> 4;              // 0: lanes 0-15, 1: lanes 16-31
    const int l16  = lane & 15;
    const int row  = m0 + l16;               // A-matrix row for this lane
    const int koff = hi * 2;                 // K offset within a 4-wide step

    // ---- cache this wave's A slice of (h + agg) in registers.
    // lane reads elements {k, k+1} with k = 4*ks + koff  ->  float2 loads
    // off a single base pointer with immediate offsets.
    v2f a_cache[K / 4];
    {
        const float2* __restrict__ hp =
            (const float2*)(h   + (size_t)row * K + koff);
        const float2* __restrict__ gp =
            (const float2*)(agg + (size_t)row * K + koff);
#pragma unroll
        for (int ks = 0; ks < K / 4; ++ks) {
            const float2 hv = hp[ks * 2];    // global_load_b64, ioffset ks*16
            const float2 gv = gp[ks * 2];
            v2f a; a.x = hv.x + gv.x; a.y = hv.y + gv.y;
            a_cache[ks] = a;
        }
    }

    for (int n0 = 0; n0 < Nout; n0 += 16) {
        const int col  = n0 + l16;
        const int bcol = (col < Nout) ? col : (Nout - 1);  // clamp pad lanes
        // B fragments: Wt[bcol][k], Wt[bcol][k+1] contiguous -> ds_load_b64
        const float2* __restrict__ bp =
            (const float2*)(ldsWt + bcol * K + koff);
        v8f c = {};
#pragma unroll
        for (int ks = 0; ks < K / 4; ++ks) {
            const float2 bv = bp[ks * 2];    // ds_load_b64, immediate offset
            v2f b; b.x = bv.x; b.y = bv.y;
            c = __builtin_amdgcn_wmma_f32_16x16x4_f32(
                    /*neg_a=*/false, a_cache[ks],
                    /*neg_b=*/false, b,
                    /*c_mod=*/(short)0, c,
                    /*reuse_a=*/false, /*reuse_b=*/false);
        }
        // ---- epilogue: fold bias (+ BN) into per-column scale/shift
        const float bb = bias[bcol];
        float scale = 1.f, shift = bb;
        if (USE_BN) {
            scale = gamma[bcol] * rsqrtf(var[bcol] + BN_EPS);
            shift = (bb - mean[bcol]) * scale + beta[bcol];
        }
        if (col < Nout) {                    // divergence only around stores
#pragma unroll
            for (int v = 0; v < 8; ++v) {
                const int m = m0 + v + 8 * hi;
                float y = c[v] * scale + shift;
                if (USE_BN) y = y > 0.f ? y : 0.f;
                out[(size_t)m * Nout + col] = y;
            }
        }
    }
}

// ----------------------------------------------------------------- driver ---
extern "C" void kernel_launch(void* const* d_in, const int* in_sizes, int n_in,
                              void* d_out, int out_size, void* d_ws, size_t ws_size,
                              hipStream_t stream) {
    (void)in_sizes; (void)n_in; (void)out_size; (void)ws_size;

    const float*     x  = (const float*)d_in[0];
    const long long* ei = (const long long*)d_in[1];     // int64 edge_index [2,E]
    const float* Ws[5], * bs[5];
    for (int i = 0; i < 5; ++i) Ws[i] = (const float*)d_in[2 + i];
    for (int i = 0; i < 5; ++i) bs[i] = (const float*)d_in[7 + i];
    const float* gam[4], * bet[4], * mu[4], * va[4];
    for (int i = 0; i < 4; ++i) gam[i] = (const float*)d_in[12 + i];
    for (int i = 0; i < 4; ++i) bet[i] = (const float*)d_in[16 + i];
    for (int i = 0; i < 4; ++i) mu[i]  = (const float*)d_in[20 + i];
    for (int i = 0; i < 4; ++i) va[i]  = (const float*)d_in[24 + i];

    float* bufA = (float*)d_ws;                           // agg scratch  [N,128]
    float* bufB = bufA + (size_t)N_NODES * F_IN;          // h ping       [N,128]
    float* bufC = bufB + (size_t)N_NODES * F_IN;          // h pong       [N,128]
    float* outF = (float*)d_out;

    const int  MT = (N_NODES / 16 + 7) / 8;               // 8 wave-tiles / block
    const dim3 GB(256);

    auto zero = [&](float* p, long long n) {
        int n4 = (int)(n / 4);
        gin_zero_kernel<<<(n4 + 255) / 256, 256, 0, stream>>>(p, n4);
    };

    // -------- Layer 0: 128 -> 64, BN + ReLU
    zero(bufA, (long long)N_NODES * F_IN);
    {
        long long tot = (long long)N_EDGES * (F_IN / 4);
        gin_scatter_kernel<F_IN><<<(int)((tot + 255) / 256), 256, 0, stream>>>(x, ei, bufA);
    }
    gin_gemm_kernel<F_IN, true><<<MT, GB, F_IN * HIDDEN * 4, stream>>>(
        x, bufA, Ws[0], bs[0], gam[0], bet[0], mu[0], va[0], bufB, N_NODES, HIDDEN);

    // -------- Layers 1..3: 64 -> 64, BN + ReLU (ping-pong bufB/bufC)
    const float* hin = bufB;
    float*       hout = bufC;
    for (int L = 1; L <= 3; ++L) {
        zero(bufA, (long long)N_NODES * HIDDEN);
        long long tot = (long long)N_EDGES * (HIDDEN / 4);
        gin_scatter_kernel<HIDDEN><<<(int)((tot + 255) / 256), 256, 0, stream>>>(hin, ei, bufA);
        gin_gemm_kernel<HIDDEN, true><<<MT, GB, HIDDEN * HIDDEN * 4, stream>>>(
            hin, bufA, Ws[L], bs[L], gam[L], bet[L], mu[L], va[L], hout, N_NODES, HIDDEN);
        float* t = (float*)hin; hin = hout; hout = t;
    }

    // -------- Layer 4: 64 -> 40, no BN, write d_out
    zero(bufA, (long long)N_NODES * HIDDEN);
    {
        long long tot = (long long)N_EDGES * (HIDDEN / 4);
        gin_scatter_kernel<HIDDEN><<<(int)((tot + 255) / 256), 256, 0, stream>>>(hin, ei, bufA);
    }
    gin_gemm_kernel<HIDDEN, false><<<MT, GB, HIDDEN * N_CLASSES * 4, stream>>>(
        hin, bufA, Ws[4], bs[4], nullptr, nullptr, nullptr, nullptr,
        outF, N_NODES, N_CLASSES);
}